// S4BlockTorch_25546465477183
// MI455X (gfx1250) — compile-verified
//
#include <hip/hip_runtime.h>
#include <hip/hip_bf16.h>

// ---------------------------------------------------------------------------
// S4 block:  LN -> causal depthwise conv (K = L = 2048) -> GELU -> Linear(+b)
//            -> + residual.    B=4, L=2048, H=512, fp32 throughout.
// ---------------------------------------------------------------------------

#define BB   4
#define LL   2048
#define HH   512
#define KK   2048
#define MM   (BB * LL)       // 8192 rows for the GEMM

typedef float v2f __attribute__((ext_vector_type(2)));
typedef float v8f __attribute__((ext_vector_type(8)));

// ---------------------------------------------------------------------------
// K1: LayerNorm.  One wave32 per row of H=512 (16 elements / lane).
// ---------------------------------------------------------------------------
__global__ void s4_ln_kernel(const float* __restrict__ x,
                             const float* __restrict__ nw,
                             const float* __restrict__ nb,
                             float* __restrict__ ln) {
  const int wave = threadIdx.x >> 5;
  const int lane = threadIdx.x & 31;
  const int row  = blockIdx.x * 8 + wave;        // [0, 8192)
  const float* xr = x + (size_t)row * HH;

  float v[16];
  float s = 0.f;
#pragma unroll
  for (int i = 0; i < 16; ++i) { v[i] = xr[lane + 32 * i]; s += v[i]; }
#pragma unroll
  for (int m = 16; m >= 1; m >>= 1) s += __shfl_xor(s, m, 32);
  const float mu = s * (1.0f / HH);

  float q = 0.f;
#pragma unroll
  for (int i = 0; i < 16; ++i) { float d = v[i] - mu; q += d * d; }
#pragma unroll
  for (int m = 16; m >= 1; m >>= 1) q += __shfl_xor(q, m, 32);
  const float inv = rsqrtf(q * (1.0f / HH) + 1e-5f);

  float* lr = ln + (size_t)row * HH;
#pragma unroll
  for (int i = 0; i < 16; ++i) {
    const int h = lane + 32 * i;
    lr[h] = (v[i] - mu) * inv * nw[h] + nb[h];
  }
}

// ---------------------------------------------------------------------------
// K2: tiled transpose  ln[b,l,h] -> ln_t[b,h,l]   (64x64 tiles, pad 65)
// ---------------------------------------------------------------------------
__global__ void s4_transpose_kernel(const float* __restrict__ ln,
                                    float* __restrict__ lnt) {
  __shared__ float s[64][65];
  const int b  = blockIdx.y;
  const int lt = blockIdx.x & 31;       // 2048/64 = 32 L-tiles
  const int ht = blockIdx.x >> 5;       //  512/64 =  8 H-tiles
  const int l0 = lt * 64, h0 = ht * 64;

  const int hi = threadIdx.x & 63;
  const int l4 = threadIdx.x >> 6;      // 0..3
#pragma unroll
  for (int i = 0; i < 16; ++i) {
    const int li = l4 + 4 * i;
    s[li][hi] = ln[((size_t)(b * LL + l0 + li)) * HH + h0 + hi];
  }
  __syncthreads();
  const int lc = threadIdx.x & 63;
  const int h4 = threadIdx.x >> 6;
#pragma unroll
  for (int i = 0; i < 16; ++i) {
    const int hr = h4 + 4 * i;
    lnt[((size_t)(b * HH + h0 + hr)) * LL + l0 + lc] = s[lc][hr];
  }
}

// ---------------------------------------------------------------------------
// K3: causal depthwise conv (full-length kernel) + bias + exact GELU.
//     One block per (b,h).  y[l] = sum_{d=0..l} wrev[d] * x[l-d].
//     Triangular tiling: jb<j tiles are unconditional, jb==j predicated.
// ---------------------------------------------------------------------------
__global__ void s4_conv_gelu_kernel(const float* __restrict__ lnt,
                                    const float* __restrict__ cw,
                                    const float* __restrict__ cb,
                                    float* __restrict__ gt) {
  __shared__ float xs[LL];
  __shared__ float wr[KK];
  const int bh  = blockIdx.x;               // [0, B*H)
  const int b   = bh >> 9;
  const int h   = bh & (HH - 1);
  const int tid = threadIdx.x;

  const float* xrow = lnt + ((size_t)(b * HH + h)) * LL;
  const float* wrow = cw + (size_t)h * KK;
#pragma unroll
  for (int i = 0; i < 8; ++i) {
    const int idx = tid + 256 * i;
    xs[idx] = xrow[idx];
    wr[idx] = wrow[KK - 1 - idx];           // reversed kernel
  }
  __syncthreads();

  float acc[8] = {0.f, 0.f, 0.f, 0.f, 0.f, 0.f, 0.f, 0.f};
#pragma unroll
  for (int jb = 0; jb < 8; ++jb) {
    const int base = 256 * jb;
#pragma unroll 4
    for (int di = 0; di < 256; ++di) {
      const float wv = wr[base + di];
      if (di <= tid) acc[jb] += wv * xs[tid - di];      // diagonal tile
#pragma unroll
      for (int j = jb + 1; j < 8; ++j)                  // full tiles
        acc[j] += wv * xs[256 * (j - jb) + tid - di];
    }
  }

  const float bias = cb[h];
  float* grow = gt + ((size_t)(b * HH + h)) * LL;
#pragma unroll
  for (int j = 0; j < 8; ++j) {
    const float y = acc[j] + bias;
    grow[256 * j + tid] = 0.5f * y * (1.0f + erff(y * 0.70710678118654752f));
  }
}

// ---------------------------------------------------------------------------
// K4: out[m,n] = sum_h gelu[m,h] * lin_w[n,h] + lin_b[n] + x[m,n]
//     fp32 WMMA (v_wmma_f32_16x16x4_f32).
//     Block tile 128(M) x 64(N), Kc = 32; 8 waves, each wave = 16x64 strip,
//     A fragment reused across 4 WMMAs (4 accumulator chains / wave).
//     LDS row strides odd (129 / 81) -> conflict-free on 64-bank LDS for
//     both the k-major cooperative stores and the lane-striped frag reads.
// ---------------------------------------------------------------------------
#define AST 129   // As row stride (128 cols + 1 pad)
#define BST 81    // Bs row stride ( 64 cols + 17 pad, odd)

__global__ void s4_gemm_kernel(const float* __restrict__ gt,
                               const float* __restrict__ lw,
                               const float* __restrict__ lb,
                               const float* __restrict__ x,
                               float* __restrict__ out) {
  __shared__ float As[32 * AST];   // [k][m], k-major
  __shared__ float Bs[32 * BST];   // [k][n]

  const int tid  = threadIdx.x;
  const int m0   = blockIdx.x * 128;        // 64 M-tiles (16 per batch)
  const int n0   = blockIdx.y * 64;         //  8 N-tiles
  const int bidx = m0 >> 11;                // batch index (m0 / 2048)
  const int l0   = m0 & (LL - 1);

  const int wave = tid >> 5, lane = tid & 31;
  const int mr   = lane & 15;               // fragment row/col within tile
  const int kof  = (lane >> 4) * 2;         // lanes 16-31 hold K=2,3

  v8f acc0 = {0.f, 0.f, 0.f, 0.f, 0.f, 0.f, 0.f, 0.f};
  v8f acc1 = acc0, acc2 = acc0, acc3 = acc0;

  const int ll = tid & 127;                 // A-load column (contiguous)
  const int hb = tid >> 7;                  // A-load row base (0..1)
  const int kk = tid & 31;                  // B-load k (coalesced in lin_w)
  const int nb = tid >> 5;                  // B-load n base (0..7)

  for (int kb = 0; kb < 16; ++kb) {
    const int h0 = kb * 32;
    __syncthreads();
    // A tile: gt[b, h0+hh, l0+ll]  (contiguous in ll) -> As[hh][ll]
#pragma unroll
    for (int i = 0; i < 16; ++i) {
      const int hh = hb + 2 * i;
      As[hh * AST + ll] =
          gt[((size_t)(bidx * HH + h0 + hh)) * LL + l0 + ll];
    }
    // B tile: lin_w[n0+nn, h0+kk]  (contiguous in kk) -> Bs[kk][nn]
#pragma unroll
    for (int i = 0; i < 8; ++i) {
      const int nn = nb + 8 * i;
      Bs[kk * BST + nn] = lw[((size_t)(n0 + nn)) * HH + h0 + kk];
    }
    __syncthreads();

#pragma unroll
    for (int ks = 0; ks < 8; ++ks) {
      const int k0 = 4 * ks + kof;
      v2f a, b0, b1, b2, b3;
      a.x  = As[k0 * AST       + wave * 16 + mr];
      a.y  = As[(k0 + 1) * AST + wave * 16 + mr];
      b0.x = Bs[k0 * BST       + mr];
      b0.y = Bs[(k0 + 1) * BST + mr];
      b1.x = Bs[k0 * BST       + 16 + mr];
      b1.y = Bs[(k0 + 1) * BST + 16 + mr];
      b2.x = Bs[k0 * BST       + 32 + mr];
      b2.y = Bs[(k0 + 1) * BST + 32 + mr];
      b3.x = Bs[k0 * BST       + 48 + mr];
      b3.y = Bs[(k0 + 1) * BST + 48 + mr];
      acc0 = __builtin_amdgcn_wmma_f32_16x16x4_f32(
          false, a, false, b0, (short)0, acc0, false, false);
      acc1 = __builtin_amdgcn_wmma_f32_16x16x4_f32(
          false, a, false, b1, (short)0, acc1, false, false);
      acc2 = __builtin_amdgcn_wmma_f32_16x16x4_f32(
          false, a, false, b2, (short)0, acc2, false, false);
      acc3 = __builtin_amdgcn_wmma_f32_16x16x4_f32(
          false, a, false, b3, (short)0, acc3, false, false);
    }
  }

  // Epilogue: D lane layout — lane&15 = N col; VGPR r = row r (+8 lanes>=16)
  const int rowOff = (lane >> 4) * 8;
#pragma unroll
  for (int r = 0; r < 8; ++r) {
    const int m = m0 + wave * 16 + rowOff + r;
    const size_t rb = (size_t)m * HH;
    const int c0 = n0 + mr;
    const int c1 = c0 + 16;
    const int c2 = c0 + 32;
    const int c3 = c0 + 48;
    out[rb + c0] = acc0[r] + lb[c0] + x[rb + c0];
    out[rb + c1] = acc1[r] + lb[c1] + x[rb + c1];
    out[rb + c2] = acc2[r] + lb[c2] + x[rb + c2];
    out[rb + c3] = acc3[r] + lb[c3] + x[rb + c3];
  }
}

// ---------------------------------------------------------------------------
// Launch:  ws layout: [0, 16MB)  = ln, later reused as g_t (gelu output)
//                     [16, 32MB) = ln_t (transposed LN)
// ---------------------------------------------------------------------------
extern "C" void kernel_launch(void* const* d_in, const int* in_sizes, int n_in,
                              void* d_out, int out_size, void* d_ws, size_t ws_size,
                              hipStream_t stream) {
  const float* x      = (const float*)d_in[0];
  const float* norm_w = (const float*)d_in[1];
  const float* norm_b = (const float*)d_in[2];
  const float* conv_w = (const float*)d_in[3];
  const float* conv_b = (const float*)d_in[4];
  const float* lin_w  = (const float*)d_in[5];
  const float* lin_b  = (const float*)d_in[6];
  float* out = (float*)d_out;

  float* ws0 = (float*)d_ws;                       // ln, then g_t
  float* ws1 = ws0 + (size_t)MM * HH;              // ln_t

  // K1: LayerNorm  (8192 rows, 8 waves/block)
  s4_ln_kernel<<<MM / 8, 256, 0, stream>>>(x, norm_w, norm_b, ws0);

  // K2: transpose ln -> ln_t
  s4_transpose_kernel<<<dim3(32 * 8, BB), 256, 0, stream>>>(ws0, ws1);

  // K3: depthwise causal conv + bias + GELU  (one block per (b,h)),
  //     writes g_t over the now-dead ln buffer.
  s4_conv_gelu_kernel<<<BB * HH, 256, 0, stream>>>(ws1, conv_w, conv_b, ws0);

  // K4: WMMA GEMM + bias + residual  (128x64 block tile, 8 waves)
  s4_gemm_kernel<<<dim3(MM / 128, HH / 64), 256, 0, stream>>>(
      ws0, lin_w, lin_b, x, out);
}